// BankedLinear_26422638805131
// MI455X (gfx1250) — compile-verified
//
#include <hip/hip_runtime.h>
#include <hip/hip_bf16.h>

// BankedLinear (MoE-style banked GEMM) for gfx1250 using V_WMMA_F32_16X16X4_F32.
//
// Shapes (fixed by the reference):
//   tensor             [B=2, T=256, IN=512]        f32
//   bank_selections    [B, T, K=2]                 i32 in [0, NB=64)
//   bank_probabilities [B, T, K]                   f32
//   weights            [NB=64, IN=512, OUT=512]    f32
//   bias               [NB, OUT]                   f32
//   out                [B, T, OUT]                 f32
//
// Strategy: bin the 1024 (token,k) rows by bank (avg 16 rows/bank == one
// 16-row WMMA M-tile), then per bank run a dense 16x512 @ 512x512 GEMM with
// fp32 WMMA, scale by probability, atomically accumulate the two k-slots.

typedef float v2f __attribute__((ext_vector_type(2)));
typedef float v8f __attribute__((ext_vector_type(8)));

#define DIM_IN   512
#define DIM_OUT  512
#define NBANKS   64
#define NTOK     512            // B*T
#define NROWS    1024           // NTOK * K
#define NSPLIT   4              // blocks per bank (each covers 8 N-tiles)

__global__ void __launch_bounds__(256) banked_zero_kernel(float* __restrict__ out, int n) {
    int i = blockIdx.x * 256 + threadIdx.x;
    if (i < n) out[i] = 0.0f;
}

__global__ void __launch_bounds__(256) banked_linear_wmma_kernel(
    const float* __restrict__ x,      // [NTOK][DIM_IN]
    const int*   __restrict__ sel,    // [NROWS]
    const float* __restrict__ prob,   // [NROWS]
    const float* __restrict__ w,      // [NBANKS][DIM_IN][DIM_OUT]
    const float* __restrict__ bias,   // [NBANKS][DIM_OUT]
    float*       __restrict__ out)    // [NTOK][DIM_OUT]
{
    __shared__ int   s_rows[NROWS];
    __shared__ float s_prob[NROWS];
    __shared__ int   s_count;
    __shared__ float s_a[16 * DIM_IN];      // 32 KB activation tile (16 rows)

    const int bank  = blockIdx.x >> 2;      // NSPLIT == 4
    const int split = blockIdx.x & 3;
    const int tid   = threadIdx.x;

    if (tid == 0) s_count = 0;
    __syncthreads();

    // ---- Bin: collect rows (token,k) that selected this bank ----
    for (int r = tid; r < NROWS; r += 256) {
        if (sel[r] == bank) {
            int p = atomicAdd(&s_count, 1);
            s_rows[p] = r;
            s_prob[p] = prob[r];
        }
    }
    __syncthreads();
    const int count = s_count;

    const int wave = tid >> 5;              // wave32: 8 waves per block
    const int lane = tid & 31;
    const int n0   = (split * 8 + wave) * 16;       // this wave's N-tile base
    const float* wb = w + (size_t)bank * (DIM_IN * DIM_OUT);

    // A-fragment addressing (ISA 7.12.2, 32-bit A 16x4):
    //   lanes 0-15  hold M=lane,    K = k0+0 (v[0]) and k0+1 (v[1])
    //   lanes 16-31 hold M=lane-16, K = k0+2 (v[0]) and k0+3 (v[1])
    const int am = lane & 15;
    const int ak = (lane >> 4) << 1;        // 0 or 2
    // B-fragment (4x16, rows striped across lanes mirroring the K split):
    //   b.x = W[k0+ak][n0+cn], b.y = W[k0+ak+1][n0+cn]
    const int cn = lane & 15;
    const float* bptr = wb + (size_t)ak * DIM_OUT + n0 + cn;

    for (int mt = 0; mt < count; mt += 16) {
        __syncthreads();   // protect s_a against waves still in previous K-loop

        // ---- Stage 16 activation rows (gathered) into LDS, 128b at a time ----
        for (int i = tid; i < 16 * (DIM_IN / 4); i += 256) {
            int m  = i >> 7;                 // / (DIM_IN/4)
            int c4 = i & 127;                // % (DIM_IN/4)
            float4 v = make_float4(0.f, 0.f, 0.f, 0.f);
            int ri = mt + m;
            if (ri < count) {
                int tok = s_rows[ri] >> 1;   // K == 2
                v = ((const float4*)(x + (size_t)tok * DIM_IN))[c4];
            }
            ((float4*)s_a)[m * (DIM_IN / 4) + c4] = v;
        }
        __syncthreads();

        // ---- GEMM: 16x512 (LDS) @ 512x16 (global, L2-resident) ----
        v8f acc = {};
        const float* ldsa = &s_a[am * DIM_IN + ak];
#pragma unroll 4
        for (int k0 = 0; k0 < DIM_IN; k0 += 4) {
            v2f a = *(const v2f*)(ldsa + k0);            // ds_load_b64
            v2f b;
            b.x = bptr[(size_t)k0 * DIM_OUT];
            b.y = bptr[(size_t)k0 * DIM_OUT + DIM_OUT];
            acc = __builtin_amdgcn_wmma_f32_16x16x4_f32(
                /*neg_a=*/false, a, /*neg_b=*/false, b,
                /*c_mod=*/(short)0, acc, /*reuse_a=*/false, /*reuse_b=*/false);
        }

        // ---- Epilogue: C layout (16x16 f32): VGPR v -> M = v + 8*(lane>=16),
        //      N = n0 + (lane&15). Add bias, scale by prob, atomic combine. ----
        const int   mbase = (lane >> 4) << 3;            // 0 or 8
        const float bv    = bias[bank * DIM_OUT + n0 + cn];
#pragma unroll
        for (int v = 0; v < 8; ++v) {
            int ri = mt + mbase + v;
            if (ri < count) {
                int   tok = s_rows[ri] >> 1;
                float p   = s_prob[ri];
                atomicAdd(out + (size_t)tok * DIM_OUT + n0 + cn, (acc[v] + bv) * p);
            }
        }
    }
}

extern "C" void kernel_launch(void* const* d_in, const int* in_sizes, int n_in,
                              void* d_out, int out_size, void* d_ws, size_t ws_size,
                              hipStream_t stream) {
    const float* x    = (const float*)d_in[0];   // tensor
    const int*   sel  = (const int*)  d_in[1];   // bank_selections
    const float* prob = (const float*)d_in[2];   // bank_probabilities
    const float* w    = (const float*)d_in[3];   // weights
    const float* bias = (const float*)d_in[4];   // bias

    float* out = (float*)d_out;                  // [NTOK][DIM_OUT] f32

    const int n = NTOK * DIM_OUT;
    banked_zero_kernel<<<(n + 255) / 256, 256, 0, stream>>>(out, n);
    banked_linear_wmma_kernel<<<NBANKS * NSPLIT, 256, 0, stream>>>(
        x, sel, prob, w, bias, out);
}